// TripletLoss_73632919323010
// MI455X (gfx1250) — compile-verified
//
#include <hip/hip_runtime.h>

typedef __attribute__((ext_vector_type(2))) float v2f;
typedef __attribute__((ext_vector_type(8))) float v8f;

#define B_N 16384
#define D_N 4096
#define C_N 10
#define MARGIN_F 0.1f

// ---------------------------------------------------------------------------
// Kernel 0: anchor squared norms. One wave per class (10 waves, 320 threads).
// ws[0..9] <- ||anchors[c]||^2
// ---------------------------------------------------------------------------
__global__ __launch_bounds__(320) void tl_norms_kernel(
    const float* __restrict__ anchors, float* __restrict__ norms) {
  int wave = threadIdx.x >> 5;
  int lane = threadIdx.x & 31;
  if (wave < C_N) {
    const float* row = anchors + (size_t)wave * D_N;
    float s = 0.0f;
    for (int i = lane; i < D_N; i += 32) {
      float v = row[i];
      s += v * v;
    }
#pragma unroll
    for (int m = 16; m >= 1; m >>= 1) s += __shfl_xor(s, m, 32);
    if (lane == 0) norms[wave] = s;
  }
}

// ---------------------------------------------------------------------------
// Kernel 1: GEMM (WMMA f32 16x16x4) + gather + per-block partial sum.
// Block = 256 threads = 8 waves = 4 row-tiles x 2 K-halves.
// Grid = 256 blocks -> 256*64 = 16384 rows exactly.
// ---------------------------------------------------------------------------
__global__ __launch_bounds__(256) void tl_main_kernel(
    const float* __restrict__ pred, const float* __restrict__ anchors,
    const int* __restrict__ clss, const int* __restrict__ negoff,
    const float* __restrict__ norms, float* __restrict__ partials) {
  __shared__ float Ctile[4][16][16];  // 4 row-tiles of S (batch-row x class)
  __shared__ float red[64];

  const int tid = threadIdx.x;

  // zero the C tiles (two K-half waves atomically combine into them)
  for (int i = tid; i < 4 * 16 * 16; i += 256) ((float*)Ctile)[i] = 0.0f;
  __syncthreads();

  const int wave = tid >> 5;
  const int lane = tid & 31;
  const int t    = wave >> 1;   // row-tile within block: 0..3
  const int kh   = wave & 1;    // K half: 0..1
  const int ln15 = lane & 15;   // A row (M) / B column (N) for this lane
  const int half = lane >> 4;   // 0: K=k,k+1   1: K=k+2,k+3

  const int row = blockIdx.x * 64 + t * 16 + ln15;       // pred row (M)
  const int cls = (ln15 < C_N) ? ln15 : (C_N - 1);        // clamp pad columns

  const float* aptr = pred    + (size_t)row * D_N + (half << 1);
  const float* bptr = anchors + (size_t)cls * D_N + (half << 1);

  v8f c0 = {};
  v8f c1 = {};

  const int k0 = kh * (D_N / 2);
  const int k1 = k0 + (D_N / 2);
  for (int k = k0; k < k1; k += 32) {  // 8 WMMAs (K=32) per iteration
    v2f av[8], bv[8];
#pragma unroll
    for (int j = 0; j < 8; ++j) {
      av[j] = *(const v2f*)(aptr + k + 4 * j);
      bv[j] = *(const v2f*)(bptr + k + 4 * j);
    }
#pragma unroll
    for (int j = 0; j < 8; j += 2) {
      c0 = __builtin_amdgcn_wmma_f32_16x16x4_f32(
          false, av[j], false, bv[j], (short)0, c0, false, false);
      c1 = __builtin_amdgcn_wmma_f32_16x16x4_f32(
          false, av[j + 1], false, bv[j + 1], (short)0, c1, false, false);
    }
  }
  v8f csum = c0 + c1;

  // C/D layout: VGPR v -> M = v + 8*half, N = ln15. Combine K-halves in LDS.
#pragma unroll
  for (int v = 0; v < 8; ++v) {
    atomicAdd(&Ctile[t][v + (half << 3)][ln15], csum[v]);  // ds_add_f32
  }
  __syncthreads();

  // Gather + per-row contribution: 64 rows per block, one thread per row.
  if (tid < 64) {
    const int tt = tid >> 4;
    const int r  = tid & 15;
    const int grow = blockIdx.x * 64 + tt * 16 + r;
    const int pos = clss[grow];
    const int neg = (pos + 1 + (negoff[grow] % 9)) % C_N;
    const float spos = Ctile[tt][r][pos];
    const float sneg = Ctile[tt][r][neg];
    red[tid] = (norms[pos] - norms[neg]) - 2.0f * (spos - sneg);
  }
  __syncthreads();

  if (tid == 0) {
    float s = 0.0f;
#pragma unroll
    for (int i = 0; i < 64; ++i) s += red[i];
    partials[blockIdx.x] = s;
  }
}

// ---------------------------------------------------------------------------
// Kernel 2: deterministic final reduction of 256 block partials.
// ---------------------------------------------------------------------------
__global__ __launch_bounds__(32) void tl_final_kernel(
    const float* __restrict__ partials, float* __restrict__ out) {
  const int lane = threadIdx.x;
  float s = 0.0f;
#pragma unroll
  for (int i = 0; i < 8; ++i) s += partials[lane + 32 * i];
#pragma unroll
  for (int m = 16; m >= 1; m >>= 1) s += __shfl_xor(s, m, 32);
  if (lane == 0) {
    out[0] = MARGIN_F + s * (1.0f / ((float)B_N * (float)D_N));
  }
}

// ---------------------------------------------------------------------------
extern "C" void kernel_launch(void* const* d_in, const int* in_sizes, int n_in,
                              void* d_out, int out_size, void* d_ws,
                              size_t ws_size, hipStream_t stream) {
  (void)in_sizes; (void)n_in; (void)out_size; (void)ws_size;
  const float* pred    = (const float*)d_in[0];
  const float* anchors = (const float*)d_in[1];
  const int*   clss    = (const int*)d_in[2];
  const int*   negoff  = (const int*)d_in[3];

  float* ws       = (float*)d_ws;
  float* norms    = ws;        // 16 floats (10 used)
  float* partials = ws + 16;   // 256 floats

  tl_norms_kernel<<<1, 320, 0, stream>>>(anchors, norms);
  tl_main_kernel<<<256, 256, 0, stream>>>(pred, anchors, clss, negoff,
                                          norms, partials);
  tl_final_kernel<<<1, 32, 0, stream>>>(partials, (float*)d_out);
}